// FlexibleGraphSAGE_42030549959220
// MI455X (gfx1250) — compile-verified
//
#include <hip/hip_runtime.h>
#include <hip/hip_bf16.h>

// ---------------------------------------------------------------------------
// GraphSAGE (3x SAGEConv + LN/ReLU/residual + L2 normalize) for gfx1250.
// GEMMs use V_WMMA_F32_16X16X4_F32 (full f32 precision; workload is
// scatter/L2-bandwidth bound so there is no reason to downcast).
// ---------------------------------------------------------------------------

#define GN    100000     // nodes
#define GE    1250000    // edges
#define GD    64         // channels

typedef float v2f __attribute__((ext_vector_type(2)));
typedef float v8f __attribute__((ext_vector_type(8)));

// ---------------------------------------------------------------------------
// Edge scatter: agg[dst] += h[src], deg[dst] += 1.  16 threads per edge,
// each thread moves one float4 chunk of the 64-float row.
// ---------------------------------------------------------------------------
__global__ __launch_bounds__(256) void sage_scatter(
    const float* __restrict__ h,
    const long long* __restrict__ src,
    const long long* __restrict__ dst,
    float* __restrict__ agg,
    float* __restrict__ deg,
    int nedges)
{
    long long idx = (long long)blockIdx.x * blockDim.x + threadIdx.x;
    int e = (int)(idx >> 4);
    int q = (int)(idx & 15);
    if (e >= nedges) return;

    int s = (int)src[e];
    int d = (int)dst[e];

    // look-ahead prefetch of a future source row (global_prefetch_b8)
    if (e + 256 < nedges) {
        int s2 = (int)src[e + 256];
        __builtin_prefetch(&h[(size_t)s2 * GD + q * 4], 0, 1);
    }

    const float4 v = *(const float4*)&h[(size_t)s * GD + q * 4];
    float* ap = &agg[(size_t)d * GD + q * 4];
    atomicAdd(ap + 0, v.x);
    atomicAdd(ap + 1, v.y);
    atomicAdd(ap + 2, v.z);
    atomicAdd(ap + 3, v.w);
    if (q == 0) atomicAdd(&deg[d], 1.0f);
}

// ---------------------------------------------------------------------------
// deg -> 1/max(deg,1) in place
// ---------------------------------------------------------------------------
__global__ __launch_bounds__(256) void sage_rdeg(float* __restrict__ deg, int n)
{
    int i = blockIdx.x * blockDim.x + threadIdx.x;
    if (i < n) deg[i] = 1.0f / fmaxf(deg[i], 1.0f);
}

// ---------------------------------------------------------------------------
// pre = (agg * rdeg) @ Wl + x @ Wr + bl      via V_WMMA_F32_16X16X4_F32.
// One wave per 16x16 output tile; 4 col-tiles per 16-row block; fused K=128.
// A layout (16x4 f32): lanes 0-15 -> K=k0,k0+1 ; lanes 16-31 -> K=k0+2,k0+3.
// C/D layout (16x16 f32): VGPR i -> M = i + 8*(lane>=16), N = lane&15.
// ---------------------------------------------------------------------------
__global__ __launch_bounds__(256) void sage_gemm_wmma(
    const float* __restrict__ agg,
    const float* __restrict__ rdeg,
    const float* __restrict__ x,
    const float* __restrict__ Wl,
    const float* __restrict__ Wr,
    const float* __restrict__ bl,
    float* __restrict__ out,
    int nrows)
{
    const int lane    = threadIdx.x & 31;
    const int wave    = threadIdx.x >> 5;
    const int tile    = blockIdx.x * 8 + wave;   // 8 waves/block
    const int rowBase = (tile >> 2) * 16;        // 4 col tiles per row block
    const int colBase = (tile & 3) * 16;
    if (rowBase >= nrows) return;

    const int mrow = rowBase + (lane & 15);      // A-row handled by this lane
    const int kHi  = (lane >> 4) * 2;            // 0 or 2 (K sub-offset)
    const int ncol = colBase + (lane & 15);      // B/D column for this lane

    const float rd = rdeg[mrow];
    v8f c = {};

    #pragma unroll
    for (int k0 = 0; k0 < GD; k0 += 4) {
        v2f a, b;
        const float2 av = *(const float2*)&agg[(size_t)mrow * GD + k0 + kHi];
        a.x = av.x * rd;
        a.y = av.y * rd;
        b.x = Wl[(k0 + kHi)     * GD + ncol];
        b.y = Wl[(k0 + kHi + 1) * GD + ncol];
        c = __builtin_amdgcn_wmma_f32_16x16x4_f32(
                false, a, false, b, (short)0, c, false, false);
    }
    #pragma unroll
    for (int k0 = 0; k0 < GD; k0 += 4) {
        v2f a, b;
        const float2 xv = *(const float2*)&x[(size_t)mrow * GD + k0 + kHi];
        a.x = xv.x;
        a.y = xv.y;
        b.x = Wr[(k0 + kHi)     * GD + ncol];
        b.y = Wr[(k0 + kHi + 1) * GD + ncol];
        c = __builtin_amdgcn_wmma_f32_16x16x4_f32(
                false, a, false, b, (short)0, c, false, false);
    }

    const float bias = bl[ncol];
    const int   mOff = (lane >> 4) * 8;
    #pragma unroll
    for (int i = 0; i < 8; ++i) {
        int r = rowBase + mOff + i;
        out[(size_t)r * GD + ncol] = c[i] + bias;
    }
}

// ---------------------------------------------------------------------------
// out = relu(LN(pre) * g + b) + res       (one wave32 per 64-wide row)
// ---------------------------------------------------------------------------
__global__ __launch_bounds__(256) void sage_ln_relu_res(
    const float* __restrict__ pre,
    const float* __restrict__ res,
    const float* __restrict__ g,
    const float* __restrict__ b,
    float* __restrict__ out,
    int nrows)
{
    int lane = threadIdx.x & 31;
    int row  = blockIdx.x * 8 + (threadIdx.x >> 5);
    if (row >= nrows) return;

    size_t base = (size_t)row * GD + lane * 2;
    float2 v = *(const float2*)&pre[base];
    float2 r = *(const float2*)&res[base];

    float s = v.x + v.y;
    #pragma unroll
    for (int m = 16; m > 0; m >>= 1) s += __shfl_xor(s, m, 32);
    float mu = s * (1.0f / 64.0f);

    float dx = v.x - mu, dy = v.y - mu;
    float q = dx * dx + dy * dy;
    #pragma unroll
    for (int m = 16; m > 0; m >>= 1) q += __shfl_xor(q, m, 32);
    float inv = rsqrtf(q * (1.0f / 64.0f) + 1e-5f);

    float2 gg = *(const float2*)&g[lane * 2];
    float2 bb = *(const float2*)&b[lane * 2];
    float2 o;
    o.x = fmaxf(dx * inv * gg.x + bb.x, 0.0f) + r.x;
    o.y = fmaxf(dy * inv * gg.y + bb.y, 0.0f) + r.y;
    *(float2*)&out[base] = o;
}

// ---------------------------------------------------------------------------
// h = pre + res ; out = h / max(||h||_2, 1e-12)   (one wave32 per row)
// ---------------------------------------------------------------------------
__global__ __launch_bounds__(256) void sage_res_l2norm(
    const float* __restrict__ pre,
    const float* __restrict__ res,
    float* __restrict__ out,
    int nrows)
{
    int lane = threadIdx.x & 31;
    int row  = blockIdx.x * 8 + (threadIdx.x >> 5);
    if (row >= nrows) return;

    size_t base = (size_t)row * GD + lane * 2;
    float2 v = *(const float2*)&pre[base];
    float2 r = *(const float2*)&res[base];
    float hx = v.x + r.x;
    float hy = v.y + r.y;

    float q = hx * hx + hy * hy;
    #pragma unroll
    for (int m = 16; m > 0; m >>= 1) q += __shfl_xor(q, m, 32);
    float inv = 1.0f / fmaxf(sqrtf(q), 1e-12f);

    float2 o;
    o.x = hx * inv;
    o.y = hy * inv;
    *(float2*)&out[base] = o;
}

// ---------------------------------------------------------------------------
// Host-side orchestration (all on `stream`, graph-capture safe)
// ---------------------------------------------------------------------------
extern "C" void kernel_launch(void* const* d_in, const int* in_sizes, int n_in,
                              void* d_out, int out_size, void* d_ws, size_t ws_size,
                              hipStream_t stream) {
    (void)in_sizes; (void)n_in; (void)out_size; (void)ws_size;

    const float*     x    = (const float*)d_in[0];
    const long long* ei   = (const long long*)d_in[1];   // int64 [2, E]
    const long long* src  = ei;
    const long long* dst  = ei + GE;
    const float* Wl[3] = { (const float*)d_in[2], (const float*)d_in[5], (const float*)d_in[8]  };
    const float* bl[3] = { (const float*)d_in[3], (const float*)d_in[6], (const float*)d_in[9]  };
    const float* Wr[3] = { (const float*)d_in[4], (const float*)d_in[7], (const float*)d_in[10] };
    const float* g[2]  = { (const float*)d_in[11], (const float*)d_in[13] };
    const float* b[2]  = { (const float*)d_in[12], (const float*)d_in[14] };

    const size_t ND = (size_t)GN * GD;
    float* ws   = (float*)d_ws;
    float* agg  = ws;            // N*D
    float* deg  = agg + ND;      // N    (becomes rdeg in place)
    float* pre  = deg + GN;      // N*D
    float* h1   = pre + ND;      // N*D
    float* h2   = h1  + ND;      // N*D
    float* outp = (float*)d_out; // N*D

    const int scatterBlocks = (int)(((long long)GE * 16 + 255) / 256); // 78125
    const int rdegBlocks    = (GN + 255) / 256;                        // 391
    const int gemmBlocks    = (GN / 16) * 4 / 8;                       // 3125
    const int rowBlocks     = GN / 8;                                  // 12500

    const float* hin[3]  = { x,  h1, h2 };
    float*       hout[2] = { h1, h2 };

    for (int layer = 0; layer < 3; ++layer) {
        hipMemsetAsync(agg, 0, ND * sizeof(float), stream);
        hipMemsetAsync(deg, 0, (size_t)GN * sizeof(float), stream);

        sage_scatter<<<scatterBlocks, 256, 0, stream>>>(
            hin[layer], src, dst, agg, deg, GE);

        sage_rdeg<<<rdegBlocks, 256, 0, stream>>>(deg, GN);

        sage_gemm_wmma<<<gemmBlocks, 256, 0, stream>>>(
            agg, deg, hin[layer], Wl[layer], Wr[layer], bl[layer], pre, GN);

        if (layer < 2) {
            sage_ln_relu_res<<<rowBlocks, 256, 0, stream>>>(
                pre, hin[layer], g[layer], b[layer], hout[layer], GN);
        } else {
            sage_res_l2norm<<<rowBlocks, 256, 0, stream>>>(
                pre, hin[layer], outp, GN);
        }
    }
}